// DN_21758304321874
// MI455X (gfx1250) — compile-verified
//
#include <hip/hip_runtime.h>
#include <math.h>

typedef float v2f __attribute__((ext_vector_type(2)));
typedef float v8f __attribute__((ext_vector_type(8)));

#define EPSF 1e-12f

// ---------------- Kernel 1: ||x||^2 ----------------
__global__ __launch_bounds__(256) void xnorm_kernel(const float* __restrict__ x, int X,
                                                    float* __restrict__ xnsq) {
  __shared__ float sm[256];
  float s = 0.f;
  for (int j = threadIdx.x; j < X; j += 256) { float v = x[j]; s += v * v; }
  sm[threadIdx.x] = s;
  __syncthreads();
  for (int st = 128; st > 0; st >>= 1) {
    if (threadIdx.x < st) sm[threadIdx.x] += sm[threadIdx.x + st];
    __syncthreads();
  }
  if (threadIdx.x == 0) xnsq[0] = sm[0];
}

// ---------------- Kernel 2: WMMA row stats for x2y_w ----------------
// One wave per 16 rows. dot(row, x_raw) via A*B (x in B column 0) and
// ||row||^2 via the Gram diagonal A*A (the f32 16x4 A-fragment layout is its
// own transpose when reused as the B operand). Tiles are DMA'd to LDS with
// async global->LDS loads (ASYNCcnt) and double-buffered: batch t+1 is issued
// before consuming batch t; async dones are in-order, so s_wait_asynccnt 8
// guarantees batch t has landed.
#define TS 68  // padded LDS stride (16B-aligned, conflict-free: bank=(4m+c)%64)
__global__ __launch_bounds__(32) void x2y_stats_kernel(const float* __restrict__ W,
                                                       const float* __restrict__ x, int X,
                                                       float* __restrict__ dotx,
                                                       float* __restrict__ nsq) {
  __shared__ float tile[2][16 * TS];
  const int lane = threadIdx.x;
  const int rb = blockIdx.x * 16;
  const int half = lane >> 4;  // 0: K0/K1, 1: K2/K3
  const int m = lane & 15;
  v8f accD = {0.f, 0.f, 0.f, 0.f, 0.f, 0.f, 0.f, 0.f};
  v8f accG = {0.f, 0.f, 0.f, 0.f, 0.f, 0.f, 0.f, 0.f};

  // Per-lane fixed (row, col) of the cooperative 16x64 tile copy; two rows of
  // 64 floats (256 B contiguous) per chunk -> fully coalesced 16 B/lane.
  unsigned ldsoff[2][8];
  const float* gbase[8];
#pragma unroll
  for (int i = 0; i < 8; ++i) {
    int flat = i * 128 + lane * 4;
    int r = flat >> 6;
    int c = flat & 63;
    // flat-LDS aperture keeps the LDS byte offset in addr[31:0]
    ldsoff[0][i] = (unsigned)(size_t)&tile[0][r * TS + c];
    ldsoff[1][i] = (unsigned)(size_t)&tile[1][r * TS + c];
    gbase[i] = W + (size_t)(rb + r) * X + c;
  }

  const int ntiles = X >> 6;
#define ISSUE_TILE(t)                                                                  \
  do {                                                                                 \
    const int _b = (t) & 1;                                                            \
    const int _k0 = (t) << 6;                                                          \
    _Pragma("unroll") for (int _i = 0; _i < 8; ++_i) {                                 \
      unsigned long long _ga = (unsigned long long)(gbase[_i] + _k0);                  \
      asm volatile("global_load_async_to_lds_b128 %0, %1, off" ::"v"(ldsoff[_b][_i]),  \
                   "v"(_ga)                                                            \
                   : "memory");                                                        \
    }                                                                                  \
  } while (0)

  ISSUE_TILE(0);
  for (int t = 0; t < ntiles; ++t) {
    if (t + 1 < ntiles) {
      ISSUE_TILE(t + 1);
      asm volatile("s_wait_asynccnt 8" ::: "memory");  // batch t landed (in-order)
    } else {
      asm volatile("s_wait_asynccnt 0" ::: "memory");
    }
    const float* tp = tile[t & 1];
    const int k0 = t << 6;
    for (int kc = 0; kc < 64; kc += 4) {
      v2f a;
      a.x = tp[m * TS + kc + 2 * half];
      a.y = tp[m * TS + kc + 2 * half + 1];
      float bx0 = 0.f, bx1 = 0.f;
      if (m == 0) {  // lanes 0 and 16 hold B column 0 (N=0): x[kc..kc+3]
        bx0 = x[k0 + kc + 2 * half];
        bx1 = x[k0 + kc + 2 * half + 1];
      }
      v2f b;
      b.x = bx0;
      b.y = bx1;
      accD = __builtin_amdgcn_wmma_f32_16x16x4_f32(false, a, false, b, (short)0, accD, false, false);
      accG = __builtin_amdgcn_wmma_f32_16x16x4_f32(false, a, false, a, (short)0, accG, false, false);
    }
  }
#undef ISSUE_TILE

  // D layout: VGPR g, lanes 0-15 -> (M=g, N=lane); lanes 16-31 -> (M=g+8, N=lane-16)
  if (lane == 0) {
    for (int i = 0; i < 8; ++i) dotx[rb + i] = accD[i];  // D[m][0], m=0..7
  }
  if (lane == 16) {
    for (int i = 0; i < 8; ++i) dotx[rb + 8 + i] = accD[i];  // D[m][0], m=8..15
  }
  for (int i = 0; i < 8; ++i) {
    if (lane == i) nsq[rb + i] = accG[i];           // Gram diag, m=0..7
    if (lane == 16 + i) nsq[rb + 8 + i] = accG[i];  // Gram G[m][m-8], m=8..15
  }
}

// ---------------- Kernel 3: z2y_w row stats ----------------
__global__ __launch_bounds__(256) void z2y_stats_kernel(const float* __restrict__ z2y,
                                                        const int* __restrict__ zptr, int Y, int Z,
                                                        float* __restrict__ zn0sq,
                                                        float* __restrict__ zd0) {
  int r = blockIdx.x * 256 + threadIdx.x;
  if (r >= Y) return;
  int zi = zptr[0];
  const float* row = z2y + (size_t)r * Z;
  float s = 0.f;
  for (int j = 0; j < Z; ++j) { float v = row[j]; s += v * v; }
  zn0sq[r] = s;
  zd0[r] = row[zi];
}

// ---------------- Kernel 4: y2z_w row norms^2 ----------------
__global__ __launch_bounds__(256) void y2z_norm_kernel(const float* __restrict__ y2z, int Y,
                                                       float* __restrict__ yzn2) {
  __shared__ float sm[256];
  const float* row = y2z + (size_t)blockIdx.x * Y;
  float s = 0.f;
  for (int j = threadIdx.x; j < Y; j += 256) { float v = row[j]; s += v * v; }
  sm[threadIdx.x] = s;
  __syncthreads();
  for (int st = 128; st > 0; st >>= 1) {
    if (threadIdx.x < st) sm[threadIdx.x] += sm[threadIdx.x + st];
    __syncthreads();
  }
  if (threadIdx.x == 0) yzn2[blockIdx.x] = sm[0];
}

// ---------------- Kernel 5: sequential train loop + finalize ----------------
// Single workgroup (1024 threads). Row updates tracked as span coefficients:
// x2y row = cu*raw_row + cx*xv ; z2y row = czu*raw_row + cz*onehot(z_idx) ;
// y2z row z_idx = g_u*raw_row + sum_k gw[k]*e_{win_k}. Coeffs stored unit-norm
// (post-l2n view), so each step is EMA-then-renormalize in closed form.
__global__ __launch_bounds__(1024) void dn_train_kernel(
    int Y, int Z, const int* __restrict__ zptr, const int* __restrict__ piptr,
    const float* __restrict__ y2z, const float* __restrict__ age_y_in,
    const float* __restrict__ age_z_in, const float* __restrict__ thr_in,
    const float* __restrict__ xnsq, const float* __restrict__ dotx_raw,
    const float* __restrict__ n0sq, const float* __restrict__ zn0sq,
    const float* __restrict__ zd0, const float* __restrict__ yzn2,
    float* __restrict__ y_pre, float* __restrict__ age_ws, float* __restrict__ thr_ws,
    float* __restrict__ out) {
  const int tid = threadIdx.x;
  const int NT = 1024;
  const int z_idx = zptr[0];
  const int per_item = piptr[0];
  const float xnc = fmaxf(sqrtf(xnsq[0]), EPSF);

  __shared__ float s_val[1024];
  __shared__ int s_idx[1024];
  __shared__ float s_aux[1024];
  __shared__ int s_nmod;
  __shared__ int s_mrow[64];
  __shared__ float s_mdx[64];
  __shared__ float s_gu;
  __shared__ int s_gwn;
  __shared__ int s_gwi[64];
  __shared__ float s_gwv[64];

  // base y_pre (post first l2n; idempotent afterwards) + working copies
  for (int r = tid; r < Y; r += NT) {
    float n0 = fmaxf(sqrtf(n0sq[r]), EPSF);
    float dx = (dotx_raw[r] / xnc) / n0;
    float zn = fmaxf(sqrtf(zn0sq[r]), EPSF);
    float dz = zd0[r] / zn;
    y_pre[r] = 0.5f * dx + 0.5f * dz;  // P_BU = P_TD = 0.5
    age_ws[r] = age_y_in[r];
    thr_ws[r] = thr_in[r];
  }
  __syncthreads();

  // thread-0 sequential state
  int mrow[64];
  float mcu[64], mcx[64], mczu[64], mcz[64];
  int nmod = 0;
  int gwi[64];
  float gwv[64];
  int gwn = 0;
  float g_u = 1.0f / fmaxf(sqrtf(yzn2[z_idx]), EPSF);  // l2n'ed original row
  float az = age_z_in[z_idx];

  for (int t = 0; t < per_item; ++t) {
    // ---- argmax(y_pre), JAX first-index tie-break ----
    float bv = -INFINITY;
    int bi = Y;
    for (int r = tid; r < Y; r += NT) {
      float v = y_pre[r];
      if (v > bv) { bv = v; bi = r; }
    }
    s_val[tid] = bv;
    s_idx[tid] = bi;
    __syncthreads();
    for (int st = NT / 2; st > 0; st >>= 1) {
      if (tid < st) {
        float ov = s_val[tid + st];
        int oi = s_idx[tid + st];
        if (ov > s_val[tid] || (ov == s_val[tid] && oi < s_idx[tid])) {
          s_val[tid] = ov;
          s_idx[tid] = oi;
        }
      }
      __syncthreads();
    }
    const int i0 = s_idx[0];
    const float r0 = s_val[0];
    __syncthreads();

    // ---- argmax(y_pre * unact) + count(unact) ----
    float bmv = -INFINITY;
    int bmi = Y;
    float cnt = 0.f;
    for (int r = tid; r < Y; r += NT) {
      float un = (age_ws[r] >= 1.0f) ? 0.0f : 1.0f;
      cnt += un;
      float v = y_pre[r] * un;
      if (v > bmv) { bmv = v; bmi = r; }
    }
    s_val[tid] = bmv;
    s_idx[tid] = bmi;
    s_aux[tid] = cnt;
    __syncthreads();
    for (int st = NT / 2; st > 0; st >>= 1) {
      if (tid < st) {
        float ov = s_val[tid + st];
        int oi = s_idx[tid + st];
        if (ov > s_val[tid] || (ov == s_val[tid] && oi < s_idx[tid])) {
          s_val[tid] = ov;
          s_idx[tid] = oi;
        }
        s_aux[tid] += s_aux[tid + st];
      }
      __syncthreads();
    }

    if (tid == 0) {
      const int alt = s_idx[0];
      const bool has_unact = s_aux[0] > 0.0f;
      const bool keep = (r0 > thr_ws[i0]) || (age_ws[i0] < 1.0f);
      const int win = keep ? i0 : (has_unact ? alt : i0);
      const float lr = 1.0f / (age_ws[win] + 1.0f);
      int e = -1;
      for (int j = 0; j < nmod; ++j)
        if (mrow[j] == win) e = j;
      float cu, cx, czu, cz;
      if (e < 0) {
        e = (nmod < 64) ? nmod++ : 63;
        mrow[e] = win;
        cu = 1.0f / fmaxf(sqrtf(n0sq[win]), EPSF);  // l2n'ed original row
        cx = 0.0f;
        czu = 1.0f / fmaxf(sqrtf(zn0sq[win]), EPSF);
        cz = 0.0f;
      } else {
        cu = mcu[e]; cx = mcx[e]; czu = mczu[e]; cz = mcz[e];
      }
      const float dxu = dotx_raw[win] / xnc;  // raw_row . x_unit
      const float zru = zd0[win];             // raw z-row value at z_idx
      // EMA on the (unit) row, then renormalize for the next step's l2n view
      float ncu = (1.0f - lr) * cu;
      float ncx = (1.0f - lr) * cx + lr;
      float nrm = fmaxf(sqrtf(fmaxf(ncu * ncu * n0sq[win] + ncx * ncx + 2.0f * ncu * ncx * dxu, 0.0f)), EPSF);
      ncu /= nrm; ncx /= nrm;
      float nczu = (1.0f - lr) * czu;
      float ncz = (1.0f - lr) * cz + lr;
      float znrm = fmaxf(sqrtf(fmaxf(nczu * nczu * zn0sq[win] + ncz * ncz + 2.0f * nczu * ncz * zru, 0.0f)), EPSF);
      nczu /= znrm; ncz /= znrm;
      mcu[e] = ncu; mcx[e] = ncx; mczu[e] = nczu; mcz[e] = ncz;
      y_pre[win] = 0.5f * (ncu * dxu + ncx) + 0.5f * (nczu * zru + ncz);
      thr_ws[win] = lr * r0 + (1.0f - lr) * thr_ws[win];
      age_ws[win] += 1.0f;
      // y2z row z_idx: EMA toward onehot(win), then renormalize
      const float zlr = 1.0f / (az + 1.0f);  // * zh[z_idx] == 1
      az += 1.0f;
      g_u *= (1.0f - zlr);
      for (int j = 0; j < gwn; ++j) gwv[j] *= (1.0f - zlr);
      int ge = -1;
      for (int j = 0; j < gwn; ++j)
        if (gwi[j] == win) ge = j;
      if (ge < 0) {
        ge = (gwn < 64) ? gwn++ : 63;
        gwi[ge] = win;
        gwv[ge] = 0.0f;
      }
      gwv[ge] += zlr;
      float nn = g_u * g_u * yzn2[z_idx];
      for (int j = 0; j < gwn; ++j) {
        float ov = y2z[(size_t)z_idx * Y + gwi[j]];
        nn += gwv[j] * gwv[j] + 2.0f * g_u * gwv[j] * ov;
      }
      float yinv = 1.0f / fmaxf(sqrtf(fmaxf(nn, 0.0f)), EPSF);
      g_u *= yinv;
      for (int j = 0; j < gwn; ++j) gwv[j] *= yinv;
    }
    __syncthreads();
  }

  // publish modified-row final x-dots and y2z coefficients
  if (tid == 0) {
    s_nmod = nmod;
    for (int j = 0; j < nmod; ++j) {
      s_mrow[j] = mrow[j];
      s_mdx[j] = mcu[j] * (dotx_raw[mrow[j]] / xnc) + mcx[j];
    }
    s_gu = g_u;
    s_gwn = gwn;
    for (int j = 0; j < gwn; ++j) { s_gwi[j] = gwi[j]; s_gwv[j] = gwv[j]; }
  }
  __syncthreads();

  // final: argmax(y_temp = x-dot * flag) + count(flag)
  float bv = -INFINITY;
  int bi = Y;
  float cnt = 0.f;
  for (int r = tid; r < Y; r += NT) {
    float flag = (age_ws[r] >= 1.0f) ? 1.0f : 0.0f;
    cnt += flag;
    float dx = (dotx_raw[r] / xnc) / fmaxf(sqrtf(n0sq[r]), EPSF);
    for (int j = 0; j < s_nmod; ++j)
      if (s_mrow[j] == r) dx = s_mdx[j];
    float v = dx * flag;
    if (v > bv) { bv = v; bi = r; }
  }
  s_val[tid] = bv;
  s_idx[tid] = bi;
  s_aux[tid] = cnt;
  __syncthreads();
  for (int st = NT / 2; st > 0; st >>= 1) {
    if (tid < st) {
      float ov = s_val[tid + st];
      int oi = s_idx[tid + st];
      if (ov > s_val[tid] || (ov == s_val[tid] && oi < s_idx[tid])) {
        s_val[tid] = ov;
        s_idx[tid] = oi;
      }
      s_aux[tid] += s_aux[tid + st];
    }
    __syncthreads();
  }
  const int jf = s_idx[0];
  if (tid < Z) {
    float val;
    if (tid == z_idx) {
      float v = s_gu * y2z[(size_t)z_idx * Y + jf];
      for (int j = 0; j < s_gwn; ++j)
        if (s_gwi[j] == jf) v += s_gwv[j];
      val = v;  // coefficients already unit-normalized (== final l2n)
    } else {
      val = y2z[(size_t)tid * Y + jf] / fmaxf(sqrtf(yzn2[tid]), EPSF);
    }
    out[tid] = val;
  }
  if (tid == 0) out[Z] = s_aux[0];  // y_activated_num
}

extern "C" void kernel_launch(void* const* d_in, const int* in_sizes, int n_in, void* d_out,
                              int out_size, void* d_ws, size_t ws_size, hipStream_t stream) {
  const float* x = (const float*)d_in[0];
  const int* z = (const int*)d_in[1];
  const int* pitem = (const int*)d_in[2];
  const float* x2y = (const float*)d_in[3];
  const float* z2y = (const float*)d_in[4];
  const float* y2z = (const float*)d_in[5];
  const float* age_y = (const float*)d_in[6];
  const float* age_z = (const float*)d_in[7];
  const float* thr = (const float*)d_in[8];
  const int X = in_sizes[0];  // 4096
  const int Y = in_sizes[8];  // 16384 (thr)
  const int Z = in_sizes[7];  // 100   (age_z)
  (void)n_in; (void)out_size; (void)ws_size;

  float* ws = (float*)d_ws;
  float* w_xnsq = ws;          // 16 (padded)
  float* w_dotx = ws + 16;     // Y
  float* w_n0sq = w_dotx + Y;  // Y
  float* w_znsq = w_n0sq + Y;  // Y
  float* w_zd0 = w_znsq + Y;   // Y
  float* w_ypre = w_zd0 + Y;   // Y
  float* w_age = w_ypre + Y;   // Y
  float* w_thr = w_age + Y;    // Y
  float* w_yzn2 = w_thr + Y;   // Z

  xnorm_kernel<<<1, 256, 0, stream>>>(x, X, w_xnsq);
  x2y_stats_kernel<<<Y / 16, 32, 0, stream>>>(x2y, x, X, w_dotx, w_n0sq);
  z2y_stats_kernel<<<(Y + 255) / 256, 256, 0, stream>>>(z2y, z, Y, Z, w_znsq, w_zd0);
  y2z_norm_kernel<<<Z, 256, 0, stream>>>(y2z, Y, w_yzn2);
  dn_train_kernel<<<1, 1024, 0, stream>>>(Y, Z, z, pitem, y2z, age_y, age_z, thr, w_xnsq,
                                          w_dotx, w_n0sq, w_znsq, w_zd0, w_yzn2, w_ypre,
                                          w_age, w_thr, (float*)d_out);
}